// VDEmbedding_23983097381329
// MI455X (gfx1250) — compile-verified
//
#include <hip/hip_runtime.h>
#include <cstdint>
#include <cstddef>

// out[r, :] = (x[r]==PAD ? 0 : mask[x[r]]) * weight[x[r], :]
// 65536 rows of 128 f32 (512B). Pure bandwidth problem (~64MB -> ~2.8us @ 23.3TB/s).

typedef float v4f __attribute__((ext_vector_type(4)));

#define ROWS            65536          // BATCH*SEQ = 16*4096
#define DIM             128            // EMBED_DIM
#define PAD_IDX         0
#define WPB             8              // waves per block (wave32 -> 256 threads)
#define THREADS         (WPB * 32)
#define TPB             4              // tiles (of WPB rows) per block, pipelined
#define ROWS_PER_BLOCK  (WPB * TPB)    // 32
#define NBLOCKS         (ROWS / ROWS_PER_BLOCK)  // 2048
#define ROW_BYTES       (DIM * 4)      // 512
#define BUF_BYTES       (WPB * ROW_BYTES) // 4096 per pipeline stage

__global__ __launch_bounds__(THREADS, 1)
void vd_embedding_kernel(const int*   __restrict__ x,
                         const float* __restrict__ weight,
                         const float* __restrict__ mask,
                         float*       __restrict__ out)
{
    // double-buffered staging: 2 x (8 rows x 512B) = 8KB LDS
    __shared__ __align__(16) float sh[2 * WPB * DIM];

    const unsigned lane = threadIdx.x & 31u;
    // force wave-uniformity so x[] / mask[] indexing scalarizes (s_load, KMcnt)
    const unsigned wv = (unsigned)__builtin_amdgcn_readfirstlane((int)(threadIdx.x >> 5));
    // generic shared pointer: low 32 bits are the LDS byte address (aperture scheme)
    const uint32_t lds_base = (uint32_t)(uintptr_t)&sh[0];

    const unsigned row0 = blockIdx.x * ROWS_PER_BLOCK + wv;  // this wave's first row

    int   tok[TPB];
    float scl[TPB];

    // Issue one async row-gather: 32 lanes x b128 = one contiguous 512B row
    // DMA'd from the (random) weight row into this wave's LDS slot.
    auto issue = [&](int t) {
        const unsigned r  = row0 + (unsigned)t * WPB;
        const int      tv = x[r];                      // wave-uniform -> scalar load
        tok[t] = tv;
        float s = mask[(unsigned)tv];                  // wave-uniform -> scalar load
        if (tv == PAD_IDX) s = 0.0f;                   // padding_idx semantics
        scl[t] = s;

        const uint64_t g = (uint64_t)(uintptr_t)(weight + ((size_t)(unsigned)tv << 7))
                         + (uint64_t)(lane << 4);
        const uint32_t l = lds_base + (uint32_t)((t & 1) * BUF_BYTES)
                         + wv * ROW_BYTES + (lane << 4);

        // ensure earlier ds_load of this buffer retired before async engine rewrites it
        asm volatile("s_wait_dscnt 0x0" ::: "memory");
        asm volatile("global_load_async_to_lds_b128 %0, %1, off"
                     :: "v"(l), "v"(g) : "memory");     // ASYNCcnt++
    };

    issue(0);
    #pragma unroll
    for (int t = 0; t < TPB; ++t) {
        if (t + 1 < TPB) {
            issue(t + 1);                               // keep 2 tiles in flight
            asm volatile("s_wait_asynccnt 0x1" ::: "memory");  // tile t landed
        } else {
            asm volatile("s_wait_asynccnt 0x0" ::: "memory");  // drain
        }

        // each wave reads back exactly the LDS region its own lanes filled
        const v4f v = *(const v4f*)(&sh[(t & 1) * (WPB * DIM) + wv * DIM + (lane << 2)]);
        const v4f o = v * scl[t];

        const unsigned r = row0 + (unsigned)t * WPB;
        // output is written once and never re-read: non-temporal so the 64MB
        // weight table stays resident in the 192MB L2 across graph replays
        __builtin_nontemporal_store(o, (v4f*)(out + ((size_t)r << 7)) + lane);
    }
}

extern "C" void kernel_launch(void* const* d_in, const int* in_sizes, int n_in,
                              void* d_out, int out_size, void* d_ws, size_t ws_size,
                              hipStream_t stream) {
    const int*   x      = (const int*)  d_in[0];   // [16, 4096] token ids
    const float* weight = (const float*)d_in[1];   // [128000, 128]
    const float* mask   = (const float*)d_in[2];   // [128000]
    float*       out    = (float*)      d_out;     // [16, 4096, 128]
    (void)in_sizes; (void)n_in; (void)out_size; (void)d_ws; (void)ws_size;

    vd_embedding_kernel<<<NBLOCKS, THREADS, 0, stream>>>(x, weight, mask, out);
}